// SVDLayer_91018946937495
// MI455X (gfx1250) — compile-verified
//
#include <hip/hip_runtime.h>
#include <math.h>

// MI455X / gfx1250, wave32. One 16x16 matrix per wave.
// Jacobi: lane = COLUMN (halves replicated), parallel round-robin pairing
// (8 disjoint pairs/round, 15 rounds/sweep). All cross-lane traffic via
// v_permlane16/v_permlanex16/v_readlane (VALU) -- no ds_bpermute.
// V never tracked: V^T = diag(1/s) * (U^T A) via WMMA.
// 12x v_wmma_f32_16x16x4_f32: W1 = U^T A, T2 = E_hat V^T, out = U T2.

typedef __attribute__((ext_vector_type(2))) float v2f;
typedef __attribute__((ext_vector_type(8))) float v8f;

// ---- lane-permute helpers (uniform compile-time patterns) --------------
__device__ __forceinline__ float pl16(float x, unsigned s1, unsigned s2) {
    unsigned xi = __float_as_uint(x);
    return __uint_as_float(
        __builtin_amdgcn_permlane16(xi, xi, s1, s2, false, false));
}
// row swap: each lane reads the same position in the opposite 16-lane row
__device__ __forceinline__ float plx16(float x) {
    unsigned xi = __float_as_uint(x);
    return __uint_as_float(__builtin_amdgcn_permlanex16(
        xi, xi, 0x76543210u, 0xFEDCBA98u, false, false));
}
__device__ __forceinline__ float bcastlane(float x, int l) {
    return __uint_as_float(__builtin_amdgcn_readlane(__float_as_uint(x), l));
}

// xor-mask permutation nibble tables (within each 16-lane row)
#define XOR1_LO 0x67452301u
#define XOR1_HI 0xEFCDAB89u
#define XOR2_LO 0x54761032u
#define XOR2_HI 0xDCFE98BAu
#define XOR4_LO 0x32107654u
#define XOR4_HI 0xBA98FEDCu
#define XOR8_LO 0xFEDCBA98u
#define XOR8_HI 0x76543210u

__device__ __forceinline__ float rsum16(float x) {
    x += pl16(x, XOR1_LO, XOR1_HI);
    x += pl16(x, XOR2_LO, XOR2_HI);
    x += pl16(x, XOR4_LO, XOR4_HI);
    x += pl16(x, XOR8_LO, XOR8_HI);
    return x;
}
__device__ __forceinline__ float rmax16(float x) {
    x = fmaxf(x, pl16(x, XOR1_LO, XOR1_HI));
    x = fmaxf(x, pl16(x, XOR2_LO, XOR2_HI));
    x = fmaxf(x, pl16(x, XOR4_LO, XOR4_HI));
    x = fmaxf(x, pl16(x, XOR8_LO, XOR8_HI));
    return x;
}

// ---- round-robin tournament schedule (compile-time) --------------------
__host__ __device__ constexpr unsigned partner_of(int c, int r) {
    if (c == 15) return (unsigned)((8 * r) % 15);
    int t = r + 15 - c;
    if (t >= 15) t -= 15;
    return (unsigned)(t == c ? 15 : t);
}
__host__ __device__ constexpr unsigned selpart(int r, int lo) {
    unsigned s = 0;
    for (int i = 0; i < 8; ++i) s |= (partner_of(lo + i, r) & 0xFu) << (4 * i);
    return s;
}
__host__ __device__ constexpr unsigned ppmask(int r) {   // bit c: c < partner
    unsigned m = 0;
    for (int i = 0; i < 16; ++i)
        if (i < (int)partner_of(i, r)) m |= 1u << i;
    return m;
}

// A/B-position 16x16x4 slice from a full per-lane register set (index = K):
// VGPR j holds K = 4*kk + j + 2*half.
__device__ __forceinline__ v2f slice4(const float (&R)[16], int kk, int half) {
    v2f a;
    a[0] = half ? R[4 * kk + 2] : R[4 * kk + 0];
    a[1] = half ? R[4 * kk + 3] : R[4 * kk + 1];
    return a;
}

// WMMA D-layout (lane = N, VGPR v holds M = v + 8*half) -> B-position slice:
// b_j(lane) = D[4kk+j+2*half][lane%16].  One row is an identity register
// read, the other fetches its register from the opposite row (permlanex16).
__device__ __forceinline__ v2f dtob(const v8f& D, int kk, int half) {
    v2f b;
#pragma unroll
    for (int j = 0; j < 2; ++j) {
        const int Mlo = 4 * kk + j;      // M needed by row 0
        const int Mhi = Mlo + 2;         // M needed by row 1
        if (Mlo < 8) {                   // both M < 8: data lives in row 0
            float own   = D[Mlo & 7];            // row 0: identity
            float cross = plx16(D[Mhi & 7]);     // row 1 pulls from row 0
            b[j] = half ? cross : own;
        } else {                         // both M >= 8: data lives in row 1
            float cross = plx16(D[Mlo & 7]);     // row 0 pulls from row 1
            float own   = D[Mhi & 7];            // row 1: identity
            b[j] = half ? own : cross;
        }
    }
    return b;
}

__global__ __launch_bounds__(256) void svdlayer_wmma_kernel(
    const float* __restrict__ x, const float* __restrict__ Wn,
    const float* __restrict__ bn, const float* __restrict__ Wt,
    const float* __restrict__ bt, float* __restrict__ out, int Bn) {

    __shared__ float ldsU[8 * 272];   // per-wave U slab, stride 17 (transpose)

    const int lane = threadIdx.x & 31;
    const int col  = lane & 15;       // column owned by this lane
    const int half = lane >> 4;
    const int wid  = threadIdx.x >> 5;
    const int wave = (int)((blockIdx.x * blockDim.x + threadIdx.x) >> 5);
    if (wave >= Bn) return;           // wave-uniform

    // ---- load column `col` of A; keep pristine copy for V^T later ------
    const float* xb = x + (size_t)wave * 256;
    float Acol[16], G[16];
#pragma unroll
    for (int r = 0; r < 16; ++r) {
        Acol[r] = xb[r * 16 + col];
        G[r] = Acol[r];
    }

    // ---- one-sided Jacobi, parallel round-robin (8 pairs / round) ------
#pragma unroll 1
    for (int sweep = 0; sweep < 8; ++sweep) {
        float nown = 0.0f;            // ||G[:,col]||^2, refreshed per sweep
#pragma unroll
        for (int r = 0; r < 16; ++r) nown = fmaf(G[r], G[r], nown);

#pragma unroll
        for (int round = 0; round < 15; ++round) {
            const unsigned ps1 = selpart(round, 0);   // const-folded
            const unsigned ps2 = selpart(round, 8);
            const unsigned pm  = ppmask(round);

            float oth[16];
#pragma unroll
            for (int r = 0; r < 16; ++r) oth[r] = pl16(G[r], ps1, ps2);
            float noth = pl16(nown, ps1, ps2);

            float cr = 0.0f;          // <g_own, g_partner>
#pragma unroll
            for (int r = 0; r < 16; ++r) cr = fmaf(G[r], oth[r], cr);

            const bool isP = (pm >> col) & 1;   // lower index plays role p
            float app = isP ? nown : noth;
            float aqq = isP ? noth : nown;
            bool doRot = (cr * cr) > (1e-24f * app * aqq + 1e-38f);
            float tau = (aqq - app) * 0.5f * __builtin_amdgcn_rcpf(cr);
            float den = fabsf(tau) + __builtin_amdgcn_sqrtf(fmaf(tau, tau, 1.0f));
            float tt  = copysignf(__builtin_amdgcn_rcpf(den), tau);
            float t_ = doRot ? tt : 0.0f;                 // branchless
            float c_ = __builtin_amdgcn_rsqf(fmaf(t_, t_, 1.0f));
            float s_ = t_ * c_;
            float sgn = isP ? -s_ : s_;  // col p: c*p - s*q ; col q: s*p + c*q
            float tsn = isP ? -t_ : t_;  // n_p -= t*apq ; n_q += t*apq
#pragma unroll
            for (int r = 0; r < 16; ++r) G[r] = fmaf(sgn, oth[r], c_ * G[r]);
            nown = fmaf(tsn, cr, nown);
        }
    }

    // ---- singular value + U column (lane-local) ------------------------
    float s2 = 0.0f;
#pragma unroll
    for (int r = 0; r < 16; ++r) s2 = fmaf(G[r], G[r], s2);
    float s_loc   = __builtin_amdgcn_sqrtf(s2);
    float inv_loc = (s2 > 1e-40f) ? __builtin_amdgcn_rcpf(s_loc) : 0.0f;
#pragma unroll
    for (int r = 0; r < 16; ++r) G[r] *= inv_loc;        // G[] = U[:,col]

    // stash U (column-major, stride 17, conflict-free) for row-layout read
#pragma unroll
    for (int r = 0; r < 16; ++r) ldsU[wid * 272 + col * 17 + r] = G[r];

    // broadcast tables via v_readlane (halves replicated -> lane j suffices)
    float skr[16], rinv[16];
#pragma unroll
    for (int j = 0; j < 16; ++j) {
        skr[j]  = bcastlane(s_loc, j);
        rinv[j] = bcastlane(inv_loc, j);
    }

    // ---- WMMA 1: W1 = U^T A  (A-pos: U^T reg-select; B-pos: Acol) ------
    v8f W1 = {0.0f, 0.0f, 0.0f, 0.0f, 0.0f, 0.0f, 0.0f, 0.0f};
#pragma unroll
    for (int kk = 0; kk < 4; ++kk) {
        v2f a = slice4(G, kk, half);     // U^T[M=k=lane][K=r]
        v2f b = slice4(Acol, kk, half);  // A[K=r][N=w=lane]
        W1 = __builtin_amdgcn_wmma_f32_16x16x4_f32(false, a, false, b,
                                                   (short)0, W1, false, false);
    }
    // W1 D-layout: lane = w, VGPR v holds k = v + 8*half.  V^T = diag(1/s)*W1.

    // ---- wv[j] = (1/s_j) * sum_w Wn[16+w] * W1[j][w] -------------------
    const float wnv = Wn[16 + col];
    float wvt[8];
#pragma unroll
    for (int v = 0; v < 8; ++v) wvt[v] = rsum16(wnv * W1[v]);
    float wvr[16];
#pragma unroll
    for (int j = 0; j < 8; ++j) wvr[j]     = rinv[j]     * bcastlane(wvt[j], 0);
#pragma unroll
    for (int j = 0; j < 8; ++j) wvr[j + 8] = rinv[j + 8] * bcastlane(wvt[j], 16);

    // wu for i = col: lane-local (Wn[r] are scalar loads)
    float wu = 0.0f;
#pragma unroll
    for (int r = 0; r < 16; ++r) wu = fmaf(Wn[r], G[r], wu);

    // ---- z[i][o] = wu[i]*S[o] + c[o]; this lane computes o = col -------
    float S = 0.0f, cv = 0.0f;
#pragma unroll
    for (int j = 0; j < 16; j += 4) {
        float4 w4 = *(const float4*)(Wt + col * 16 + j);
        S += w4.x + w4.y + w4.z + w4.w;
        cv = fmaf(wvr[j + 0], w4.x, cv);
        cv = fmaf(wvr[j + 1], w4.y, cv);
        cv = fmaf(wvr[j + 2], w4.z, cv);
        cv = fmaf(wvr[j + 3], w4.w, cv);
    }
    float cterm = cv + bn[0] * S + bt[col];

    float z[16];                         // layout: lane = i, reg = o
#pragma unroll
    for (int o = 0; o < 16; ++o)
        z[o] = fmaf(wu, bcastlane(S, o), bcastlane(cterm, o));

    // ---- both softmaxes from one set of exponentials (global max) ------
    float mx = z[0];
#pragma unroll
    for (int o = 1; o < 16; ++o) mx = fmaxf(mx, z[o]);
    float mg = rmax16(mx);               // global max: stabilizes both
    float e[16];
    float se = 0.0f;
#pragma unroll
    for (int o = 0; o < 16; ++o) { e[o] = __expf(z[o] - mg); se += e[o]; }
    float ra = __builtin_amdgcn_rcpf(se);
    float beta[16];
#pragma unroll
    for (int o = 0; o < 16; ++o) {
        z[o]    = e[o] * ra;                                     // alpha
        beta[o] = e[o] * __builtin_amdgcn_rcpf(rsum16(e[o]));    // beta
    }

    // E_hat[k=lane][j=reg] = s_k * alpha + beta * s_j
    float E[16];
#pragma unroll
    for (int j = 0; j < 16; ++j) E[j] = fmaf(s_loc, z[j], beta[j] * skr[j]);

    // ---- WMMA 2: T2 = E_hat @ V^T --------------------------------------
    v8f VT;
#pragma unroll
    for (int v = 0; v < 8; ++v) VT[v] = W1[v] * (half ? rinv[v + 8] : rinv[v]);
    v8f T2 = {0.0f, 0.0f, 0.0f, 0.0f, 0.0f, 0.0f, 0.0f, 0.0f};
#pragma unroll
    for (int kk = 0; kk < 4; ++kk) {
        v2f a = slice4(E, kk, half);     // E[M=k=lane][K=j]
        v2f b = dtob(VT, kk, half);      // V^T[K=j][N=w]
        T2 = __builtin_amdgcn_wmma_f32_16x16x4_f32(false, a, false, b,
                                                   (short)0, T2, false, false);
    }

    // row-layout U from LDS (same-wave DS ops in-order; no barrier needed)
    float Urow[16];
#pragma unroll
    for (int j = 0; j < 16; ++j) Urow[j] = ldsU[wid * 272 + j * 17 + col];

    // ---- WMMA 3: out = U @ T2 ------------------------------------------
    v8f OT = {0.0f, 0.0f, 0.0f, 0.0f, 0.0f, 0.0f, 0.0f, 0.0f};
#pragma unroll
    for (int kk = 0; kk < 4; ++kk) {
        v2f a = slice4(Urow, kk, half);  // U[M=r=lane][K=k]
        v2f b = dtob(T2, kk, half);      // T2[K=k][N=w]
        OT = __builtin_amdgcn_wmma_f32_16x16x4_f32(false, a, false, b,
                                                   (short)0, OT, false, false);
    }

    // OT D-layout: lane holds out[r = v + 8*half][w = col]; coalesced stores
    float* ob = out + (size_t)wave * 256;
#pragma unroll
    for (int v = 0; v < 8; ++v) ob[(v + 8 * half) * 16 + col] = OT[v];
}

extern "C" void kernel_launch(void* const* d_in, const int* in_sizes, int n_in,
                              void* d_out, int out_size, void* d_ws, size_t ws_size,
                              hipStream_t stream) {
    const float* x  = (const float*)d_in[0];
    const float* Wn = (const float*)d_in[1];
    const float* bn = (const float*)d_in[2];
    const float* Wt = (const float*)d_in[3];
    const float* bt = (const float*)d_in[4];
    float* out = (float*)d_out;

    const int Bn = in_sizes[0] / 256;    // 65536 matrices
    const int wavesPerBlock = 8;         // 256 threads = 8 wave32
    const int threads = wavesPerBlock * 32;
    const int blocks = (Bn + wavesPerBlock - 1) / wavesPerBlock;
    svdlayer_wmma_kernel<<<blocks, threads, 0, stream>>>(x, Wn, bn, Wt, bt, out, Bn);
}